// FastGTLayer_60309930770875
// MI455X (gfx1250) — compile-verified
//
#include <hip/hip_runtime.h>
#include <hip/hip_bf16.h>
#include <stdint.h>

// Problem constants (match reference): C,T,N,E,D
#define C_ 2
#define T_ 4
#define N_ 50000
#define E_ 400000
#define D_ 64
#define M_ (T_ * E_)        // 1,600,000 flattened edges
#define TILE 256
#define NBLK (M_ / TILE)    // 6250 (exact)

// LDS byte offset of a shared-memory pointer (generic -> addrspace(3) cast).
__device__ __forceinline__ uint32_t lds_addr_u32(const void* p) {
    return (uint32_t)(uintptr_t)(__attribute__((address_space(3))) const void*)p;
}

// CDNA5 async global -> LDS copy (tracked by ASYNCcnt). GV addressing:
// 64-bit global address in a VGPR pair, LDS destination byte-address in a VGPR.
__device__ __forceinline__ void async_load_b32(uint32_t lds, const void* gaddr) {
    asm volatile("global_load_async_to_lds_b32 %0, %1, off"
                 :
                 : "v"(lds), "v"(gaddr)
                 : "memory");
}

__global__ __launch_bounds__(256) void gt_scatter_kernel(
    const float* __restrict__ H,       // [C, N, D]
    const int*   __restrict__ eidx,    // [T, 2, E]
    const float* __restrict__ evals,   // [T, E]  (flat index == flattened m)
    const float* __restrict__ weight,  // [C, T]
    float*       __restrict__ out)     // [C, N, D] (pre-zeroed)
{
    __shared__ int   s_row[TILE];
    __shared__ int   s_col[TILE];
    __shared__ float s_val[TILE];

    const int tid  = threadIdx.x;      // 0..255 (8 waves of 32)
    const int wave = tid >> 5;
    const int lane = tid & 31;
    const int m0   = blockIdx.x * TILE;

    // ---- stage this block's edge tile into LDS via async tensor path ----
    {
        const int m = m0 + tid;
        const int t = m / E_;                    // edge type
        const int e = m - t * E_;
        const int*   rp = eidx + (size_t)t * (2 * E_) + e;       // edge_index[t,0,e]
        const int*   cp = rp + E_;                               // edge_index[t,1,e]
        const float* vp = evals + m;                             // edge_values flat
        async_load_b32(lds_addr_u32(&s_row[tid]), rp);
        async_load_b32(lds_addr_u32(&s_col[tid]), cp);
        async_load_b32(lds_addr_u32(&s_val[tid]), vp);
    }

    // ---- softmax(weight, axis=1): 8 cache-hot scalars, computed per thread ----
    float filt[C_][T_];
#pragma unroll
    for (int c = 0; c < C_; ++c) {
        float w[T_];
        float mx = -1e30f;
#pragma unroll
        for (int t = 0; t < T_; ++t) { w[t] = weight[c * T_ + t]; mx = fmaxf(mx, w[t]); }
        float s = 0.f;
#pragma unroll
        for (int t = 0; t < T_; ++t) { w[t] = __expf(w[t] - mx); s += w[t]; }
        const float inv = 1.0f / s;
#pragma unroll
        for (int t = 0; t < T_; ++t) filt[c][t] = w[t] * inv;
    }

    // Async copies must land in LDS before any wave reads the tile.
    asm volatile("s_wait_asynccnt 0" ::: "memory");
    __syncthreads();

    // ---- 2*TILE (channel, edge) tasks; one wave32 per task, float2 per lane ----
#pragma unroll 1
    for (int task = wave; task < C_ * TILE; task += 8) {
        const int   j  = task >> 1;
        const int   c  = task & 1;
        const int   r  = s_row[j];
        const int   cl = s_col[j];
        const float v  = s_val[j];
        const int   t  = (m0 + j) / E_;
        const float wv = filt[c][t] * v;

        const float2* hp = (const float2*)(H + ((size_t)c * N_ + cl) * (size_t)D_);
        const float2  h  = hp[lane];                      // global_load_b64, coalesced

        float* op = out + ((size_t)c * N_ + r) * (size_t)D_ + lane * 2;
        __hip_atomic_fetch_add(op + 0, wv * h.x, __ATOMIC_RELAXED, __HIP_MEMORY_SCOPE_AGENT);
        __hip_atomic_fetch_add(op + 1, wv * h.y, __ATOMIC_RELAXED, __HIP_MEMORY_SCOPE_AGENT);
    }
}

extern "C" void kernel_launch(void* const* d_in, const int* in_sizes, int n_in,
                              void* d_out, int out_size, void* d_ws, size_t ws_size,
                              hipStream_t stream) {
    const float* H      = (const float*)d_in[0];   // [C,N,D] f32
    const int*   eidx   = (const int*)d_in[1];     // [T,2,E] i32
    const float* evals  = (const float*)d_in[2];   // [T,E]   f32
    const float* weight = (const float*)d_in[3];   // [C,T]   f32
    // d_in[4] = num_nodes (compile-time constant here)
    float* out = (float*)d_out;

    hipMemsetAsync(out, 0, (size_t)C_ * N_ * D_ * sizeof(float), stream);
    gt_scatter_kernel<<<NBLK, 256, 0, stream>>>(H, eidx, evals, weight, out);
}